// SsimLoss_34179349741945
// MI455X (gfx1250) — compile-verified
//
#include <hip/hip_runtime.h>
#include <math.h>

typedef float v2f __attribute__((ext_vector_type(2)));
typedef float v8f __attribute__((ext_vector_type(8)));

#define HW    512
#define SLICE (HW * HW)          // 262144 = 2^18
#define TPS   1024               // (512/16)^2 tiles per slice

// cv2.getGaussianKernel(11, 1.5), normalized (computed in double, rounded to f32)
__device__ __constant__ float G11[11] = {
    0.00102841f, 0.00759875f, 0.03600072f, 0.10936095f, 0.21300556f, 0.26601172f,
    0.21300556f, 0.10936095f, 0.03600072f, 0.00759875f, 0.00102841f
};

// ---------------------------------------------------------------------------
// Depth mean: b2[b,h,w] = mean_d img2[b,d,h,w]
// ---------------------------------------------------------------------------
__global__ void depth_mean_kernel(const float* __restrict__ img2,
                                  float* __restrict__ b2, int n) {
    int i = blockIdx.x * blockDim.x + threadIdx.x;
    if (i >= n) return;
    int b  = i >> 18;              // / SLICE
    int hw = i & (SLICE - 1);
    const float* p = img2 + (size_t)b * 16 * SLICE + hw;
    float s = 0.f;
#pragma unroll
    for (int d = 0; d < 16; ++d) s += p[(size_t)d * SLICE];
    b2[i] = s * (1.f / 16.f);
}

// ---------------------------------------------------------------------------
// Per-tile SSIM via FP32 WMMA (V_WMMA_F32_16X16X4_F32).
// One wave32 per 16x16 output tile. Separable 11-tap Gaussian blur expressed
// as banded matmuls, K split into 7 chunks of 4 (K=26 padded to 28).
// ---------------------------------------------------------------------------
__global__ __launch_bounds__(32) void ssim_tile_kernel(const float* __restrict__ A,
                                                       const float* __restrict__ B,
                                                       float* __restrict__ partials) {
    __shared__ float sA[26][28];        // a halo tile (cols 26,27 = K padding)
    __shared__ float sB[26][28];        // b halo tile
    __shared__ float sH[5][28][16];     // h-blurred {a,b,aa,bb,ab} (rows 26,27 pad)

    const int bx   = blockIdx.x;
    const int s    = bx >> 10;          // slice index
    const int t    = bx & (TPS - 1);
    const int tr   = (t >> 5) << 4;     // tile row origin
    const int tc   = (t & 31) << 4;     // tile col origin
    const int lane = threadIdx.x;
    const int hi   = lane >> 4;         // lane half (0 or 1)
    const int lo   = lane & 15;

    const float* As = A + (size_t)s * SLICE;
    const float* Bs = B + (size_t)s * SLICE;

    // --- stage zero-padded 26x26 halos (plus 2 zero K-pad columns) ----------
    for (int idx = lane; idx < 26 * 28; idx += 32) {
        int i = idx / 28, j = idx - i * 28;
        int r = tr - 5 + i, c = tc - 5 + j;
        bool ok = (j < 26) && (r >= 0) && (r < HW) && (c >= 0) && (c < HW);
        float va = 0.f, vb = 0.f;
        if (ok) { va = As[r * HW + c]; vb = Bs[r * HW + c]; }
        sA[i][j] = va;
        sB[i][j] = vb;
    }
    // zero the two K-padding rows of sH (their weights are 0, but avoid NaN*0)
    for (int idx = lane; idx < 5 * 2 * 16; idx += 32) {
        int q = idx >> 5, rem = idx & 31;
        sH[q][26 + (rem >> 4)][rem & 15] = 0.f;
    }
    __syncthreads();

    // --- banded Gaussian weight fragments ----------------------------------
    // A(16x4) layout: lane: M=lo, K = q*4 + 2*hi + v.   B(4x16): K = q*4 + v + 2*hi, N = lo.
    // Both passes use weight(k, n) = G[k - n] for k-n in [0,10] -> identical fragment.
    v2f wf[7];
#pragma unroll
    for (int q = 0; q < 7; ++q) {
        int k0 = q * 4 + 2 * hi;
        int d0 = k0 - lo, d1 = k0 + 1 - lo;
        int c0 = d0 < 0 ? 0 : (d0 > 10 ? 10 : d0);
        int c1 = d1 < 0 ? 0 : (d1 > 10 ? 10 : d1);
        wf[q][0] = (d0 >= 0 && d0 <= 10) ? G11[c0] : 0.f;
        wf[q][1] = (d1 >= 0 && d1 <= 10) ? G11[c1] : 0.f;
    }

    const v8f vzero = {0.f, 0.f, 0.f, 0.f, 0.f, 0.f, 0.f, 0.f};

    // --- horizontal pass: Hout(26x16) = In(26x26) x Wh(26x16), two 16-row groups
#pragma unroll
    for (int rg = 0; rg < 2; ++rg) {
        const int rbase = rg ? 10 : 0;  // rows 0..15 then 10..25 (overlap = same values)
        v8f acc[5];
#pragma unroll
        for (int qy = 0; qy < 5; ++qy) acc[qy] = vzero;

#pragma unroll
        for (int q = 0; q < 7; ++q) {
            const int row = rbase + lo;
            const int k0  = q * 4 + 2 * hi;
            float a0 = sA[row][k0], a1 = sA[row][k0 + 1];
            float b0 = sB[row][k0], b1 = sB[row][k0 + 1];
            v2f fr[5];
            fr[0][0] = a0;      fr[0][1] = a1;
            fr[1][0] = b0;      fr[1][1] = b1;
            fr[2][0] = a0 * a0; fr[2][1] = a1 * a1;
            fr[3][0] = b0 * b0; fr[3][1] = b1 * b1;
            fr[4][0] = a0 * b0; fr[4][1] = a1 * b1;
#pragma unroll
            for (int qy = 0; qy < 5; ++qy)
                acc[qy] = __builtin_amdgcn_wmma_f32_16x16x4_f32(
                    false, fr[qy], false, wf[q], (short)0, acc[qy], false, false);
        }
        // D layout: vgpr v -> row rbase + v + 8*hi, col lo
#pragma unroll
        for (int qy = 0; qy < 5; ++qy)
#pragma unroll
            for (int v = 0; v < 8; ++v)
                sH[qy][rbase + v + 8 * hi][lo] = acc[qy][v];
    }
    __syncthreads();

    // --- vertical pass: Vout(16x16) = Wv(16x26) x Hout(26x16) ---------------
    v8f vacc[5];
#pragma unroll
    for (int qy = 0; qy < 5; ++qy) vacc[qy] = vzero;
#pragma unroll
    for (int q = 0; q < 7; ++q) {
        const int k0 = q * 4 + 2 * hi;   // B layout: K = q*4 + v + 2*hi, N = lo
#pragma unroll
        for (int qy = 0; qy < 5; ++qy) {
            v2f bf;
            bf[0] = sH[qy][k0][lo];
            bf[1] = sH[qy][k0 + 1][lo];
            vacc[qy] = __builtin_amdgcn_wmma_f32_16x16x4_f32(
                false, wf[q], false, bf, (short)0, vacc[qy], false, false);
        }
    }

    // --- SSIM formula + tile reduction --------------------------------------
    const float C1 = 1e-4f, C2 = 9e-4f;
    float sum = 0.f;
#pragma unroll
    for (int i = 0; i < 8; ++i) {
        float mu1 = vacc[0][i], mu2 = vacc[1][i];
        float baa = vacc[2][i], bbb = vacc[3][i], bab = vacc[4][i];
        float mu1s = mu1 * mu1, mu2s = mu2 * mu2, mu12 = mu1 * mu2;
        float s1 = baa - mu1s, s2 = bbb - mu2s, s12 = bab - mu12;
        float lum = (2.f * mu12 + C1) / (mu1s + mu2s + C1);
        float con = (2.f * s12 + C2) / (s1 + s2 + C2);
        float str = (s12 + C2) / (sqrtf(s1) * sqrtf(s2) + C2);
        sum += lum * con * str;
    }
#pragma unroll
    for (int off = 16; off >= 1; off >>= 1)
        sum += __shfl_xor(sum, off, 32);
    if (lane == 0) partials[bx] = sum;
}

// ---------------------------------------------------------------------------
// Deterministic final reduction: loss = (1 - mean3d) + (1 - mean2d)
// ---------------------------------------------------------------------------
__global__ __launch_bounds__(256) void final_reduce_kernel(const float* __restrict__ p3, int n3,
                                                           const float* __restrict__ p2, int n2,
                                                           float* __restrict__ out) {
    __shared__ float sh[256];
    const int tid = threadIdx.x;
    float t3 = 0.f, t2 = 0.f;
    for (int i = tid; i < n3; i += 256) t3 += p3[i];
    for (int i = tid; i < n2; i += 256) t2 += p2[i];

    sh[tid] = t3; __syncthreads();
    for (int off = 128; off >= 1; off >>= 1) {
        if (tid < off) sh[tid] += sh[tid + off];
        __syncthreads();
    }
    float s3 = sh[0]; __syncthreads();

    sh[tid] = t2; __syncthreads();
    for (int off = 128; off >= 1; off >>= 1) {
        if (tid < off) sh[tid] += sh[tid + off];
        __syncthreads();
    }
    if (tid == 0) {
        float s2 = sh[0];
        float m3 = s3 / (64.0f * (float)SLICE);
        float m2 = s2 / (4.0f * (float)SLICE);
        out[0] = (1.f - m3) + (1.f - m2);
    }
}

// ---------------------------------------------------------------------------
extern "C" void kernel_launch(void* const* d_in, const int* in_sizes, int n_in,
                              void* d_out, int out_size, void* d_ws, size_t ws_size,
                              hipStream_t stream) {
    const float* img1_3d = (const float*)d_in[0];   // [4,1,16,512,512]
    const float* img1_2d = (const float*)d_in[1];   // [4,1,512,512]
    const float* img2    = (const float*)d_in[2];   // [4,1,16,512,512]
    float* out = (float*)d_out;

    // workspace layout (floats): b2 [1M] | partials3d [64K] | partials2d [4K]
    float* wsf = (float*)d_ws;
    float* b2  = wsf;
    float* p3  = wsf + (1 << 20);
    float* p2  = p3 + (64 * TPS);

    const int n2d = 4 * SLICE;
    depth_mean_kernel<<<(n2d + 255) / 256, 256, 0, stream>>>(img2, b2, n2d);

    ssim_tile_kernel<<<64 * TPS, 32, 0, stream>>>(img1_3d, img2, p3);   // 3D part
    ssim_tile_kernel<<< 4 * TPS, 32, 0, stream>>>(img1_2d, b2,  p2);    // 2D part

    final_reduce_kernel<<<1, 256, 0, stream>>>(p3, 64 * TPS, p2, 4 * TPS, out);
}